// RGCNLayer_23467701305774
// MI455X (gfx1250) — compile-verified
//
#include <hip/hip_runtime.h>

typedef float v2f __attribute__((ext_vector_type(2)));
typedef float v8f __attribute__((ext_vector_type(8)));

#define N_NODES_C 50000
#define N_EDGES_C 600000
#define DIM 128
#define N_BASES_C 4
#define N_REL_C 5          // relation count (edge types)
#define N_MATS 6           // W0 + 5 relations
#define M_TILES (N_NODES_C / 16)   // 3125, exact

// ---------------------------------------------------------------------------
// Kernel 1: Wall[0] = W0 ; Wall[1+r] = sum_k A[r,k] * B[k]   (each 128x128)
// ---------------------------------------------------------------------------
__global__ __launch_bounds__(256) void build_weights(
    const float* __restrict__ B, const float* __restrict__ A,
    const float* __restrict__ W0, float* __restrict__ Wall) {
    const int MAT = DIM * DIM;
    int tid = blockIdx.x * blockDim.x + threadIdx.x;
    if (tid >= N_MATS * MAT) return;
    int rel = tid / MAT;
    int idx = tid - rel * MAT;
    float v;
    if (rel == 0) {
        v = W0[idx];
    } else {
        const float* a = A + (rel - 1) * N_BASES_C;
        v = a[0] * B[idx] + a[1] * B[MAT + idx] +
            a[2] * B[2 * MAT + idx] + a[3] * B[3 * MAT + idx];
    }
    Wall[tid] = v;
}

// ---------------------------------------------------------------------------
// Kernel 2: per block: one 16-row node tile; 6 waves = 6 weight matrices.
// Wave computes D[16x128] = Xtile[16x128] * W^T via v_wmma_f32_16x16x4_f32.
// rel 0 -> d_out (self-loop term), rel 1..5 -> H[rel-1] in workspace.
// ---------------------------------------------------------------------------
__global__ __launch_bounds__(192) void gemm_wmma(
    const float* __restrict__ x, const float* __restrict__ Wall,
    float* __restrict__ out, float* __restrict__ H) {
    __shared__ float xt[16 * DIM];   // 8 KB x tile shared by all 6 waves

    const int mtile = blockIdx.x;
    for (int i = threadIdx.x; i < 16 * DIM; i += blockDim.x)
        xt[i] = x[(size_t)mtile * 16 * DIM + i];
    __syncthreads();

    const int wave = threadIdx.x >> 5;      // 0..5 == which weight matrix
    const int lane = threadIdx.x & 31;
    const int half = lane >> 4;             // 0: K+0..1 / 1: K+2..3
    const int l15  = lane & 15;

    const float* W = Wall + (size_t)wave * DIM * DIM;
    float* dst = (wave == 0)
        ? (out + (size_t)mtile * 16 * DIM)
        : (H + (size_t)(wave - 1) * N_NODES_C * DIM + (size_t)mtile * 16 * DIM);

    v8f acc[8] = {};   // 8 N-tiles of 16x16 f32 accumulators

    const float* arow = xt + l15 * DIM + half * 2;   // A: row M=l15, pair at K
#pragma unroll 4
    for (int kb = 0; kb < 32; ++kb) {
        const int k = kb * 4;
        v2f a = *(const v2f*)(arow + k);
#pragma unroll
        for (int nt = 0; nt < 8; ++nt) {
            // B(k,n) = W[n][k] : lane l15 holds column n = nt*16+l15, pair at K
            v2f b = *(const v2f*)(W + (size_t)(nt * 16 + l15) * DIM + k + half * 2);
            acc[nt] = __builtin_amdgcn_wmma_f32_16x16x4_f32(
                false, a, false, b, (short)0, acc[nt], false, false);
        }
    }

    // C/D layout: VGPR j -> row M = j + 8*half, col N = l15 (within N-tile)
#pragma unroll
    for (int nt = 0; nt < 8; ++nt) {
#pragma unroll
        for (int j = 0; j < 8; ++j) {
            dst[(size_t)(half * 8 + j) * DIM + nt * 16 + l15] = acc[nt][j];
        }
    }
}

// ---------------------------------------------------------------------------
// Kernel 3: one wave per edge. Gather 512B row of H[type][src] (L2-resident),
// atomically add into out[tgt]. Each lane handles 4 floats.
// ---------------------------------------------------------------------------
__global__ __launch_bounds__(256) void scatter_edges(
    const int* __restrict__ edge_index, const int* __restrict__ edge_type,
    const float* __restrict__ H, float* __restrict__ out) {
    const int wid = blockIdx.x * (blockDim.x >> 5) + (threadIdx.x >> 5);
    if (wid >= N_EDGES_C) return;
    const int lane = threadIdx.x & 31;

    const int src = edge_index[wid];
    const int tgt = edge_index[N_EDGES_C + wid];
    const int ty  = edge_type[wid];

    const float4 v = *(const float4*)(
        H + ((size_t)ty * N_NODES_C + (size_t)src) * DIM + lane * 4);
    float* o = out + (size_t)tgt * DIM + lane * 4;

    __hip_atomic_fetch_add(o + 0, v.x, __ATOMIC_RELAXED, __HIP_MEMORY_SCOPE_AGENT);
    __hip_atomic_fetch_add(o + 1, v.y, __ATOMIC_RELAXED, __HIP_MEMORY_SCOPE_AGENT);
    __hip_atomic_fetch_add(o + 2, v.z, __ATOMIC_RELAXED, __HIP_MEMORY_SCOPE_AGENT);
    __hip_atomic_fetch_add(o + 3, v.w, __ATOMIC_RELAXED, __HIP_MEMORY_SCOPE_AGENT);
}

// ---------------------------------------------------------------------------
extern "C" void kernel_launch(void* const* d_in, const int* in_sizes, int n_in,
                              void* d_out, int out_size, void* d_ws, size_t ws_size,
                              hipStream_t stream) {
    const float* x  = (const float*)d_in[0];
    const int* ei   = (const int*)d_in[1];   // [2, E]
    const int* et   = (const int*)d_in[2];   // [E]
    const float* B  = (const float*)d_in[3]; // [4,128,128]
    const float* A  = (const float*)d_in[4]; // [5,4]
    const float* W0 = (const float*)d_in[5]; // [128,128]
    float* out = (float*)d_out;

    // Workspace layout: Wall[6*128*128] | H[5 * N * 128]
    float* Wall = (float*)d_ws;
    float* H    = Wall + (size_t)N_MATS * DIM * DIM;

    const int wtot = N_MATS * DIM * DIM;                 // 98304
    build_weights<<<(wtot + 255) / 256, 256, 0, stream>>>(B, A, W0, Wall);

    gemm_wmma<<<M_TILES, 192, 0, stream>>>(x, Wall, out, H);

    const int blocks = N_EDGES_C / 8;                    // 8 waves (edges) / block
    scatter_edges<<<blocks, 256, 0, stream>>>(ei, et, H, out);
}